// MYGCN_12489764897162
// MI455X (gfx1250) — compile-verified
//
#include <hip/hip_runtime.h>

typedef __attribute__((ext_vector_type(2))) float v2f;
typedef __attribute__((ext_vector_type(8))) float v8f;

#define NN 100000
#define FF 256
#define EE 1600000
#define HH1 128
#define HH2 64

#define GEMM_WAVES 4

// ---------------------------------------------------------------------------
// zero fill (graph-capture safe, deterministic init of d_ws / d_out regions)
// ---------------------------------------------------------------------------
__global__ void gcn_zero(float* __restrict__ p, int n) {
    int i = blockIdx.x * blockDim.x + threadIdx.x;
    if (i < n) p[i] = 0.0f;
}

// relu in place
__global__ void gcn_relu(float* __restrict__ p, int n) {
    int i = blockIdx.x * blockDim.x + threadIdx.x;
    if (i < n) p[i] = fmaxf(p[i], 0.0f);
}

// ---------------------------------------------------------------------------
// Dual-support WMMA fp32 GEMM (compile-time K, Nout):
//   C0[M,Nout] = A[M,K] @ B0[K,Nout]
//   C1[M,Nout] = A[M,K] @ B1[K,Nout]
// - A-fragment loaded once per k-step, fed to both supports' WMMA chains.
// - 4 k-steps per iteration -> 8 independent accumulators: RAW reuse distance
//   of 8 WMMAs (>> ISA 7.12.1 hazard window), loads batch into clauses.
// - All intra-iteration offsets are compile-time -> load-immediate offsets,
//   only 3 pointer bumps per iteration.
//
// Fragment layouts per CDNA5 ISA 7.12.2 (wave32):
//   A 16x4:  lane l: m = l&15, half = l>>4; a[v] = A[m][2*half + v]
//   B 4x16:  lane l: n = l&15, half = l>>4; b[v] = B[2*half + v][n]
//   C 16x16: lane l, vgpr v: row = v + 8*(l>>4), col = l&15
// ---------------------------------------------------------------------------
template <int K, int Nout>
__global__ __launch_bounds__(32 * GEMM_WAVES)
void gcn_gemm_wmma2(const float* __restrict__ A,
                    const float* __restrict__ B0, const float* __restrict__ B1,
                    float* __restrict__ C0, float* __restrict__ C1, int M) {
    static_assert(K % 16 == 0 && Nout % 16 == 0, "tile dims");
    const int lane = threadIdx.x;          // 0..31
    const int half = lane >> 4;            // 0 or 1
    const int mn   = lane & 15;
    const int tileN = blockIdx.x;
    const int tileM = blockIdx.y * GEMM_WAVES + threadIdx.y;
    if (tileM * 16 >= M) return;           // wave-uniform: EXEC stays all-1s

    const float* __restrict__ ap  = A + (size_t)(tileM * 16 + mn) * K + 2 * half;
    const size_t boff = (size_t)(2 * half) * Nout + tileN * 16 + mn;
    const float* __restrict__ b0c = B0 + boff;
    const float* __restrict__ b1c = B1 + boff;

    v8f s0a = {}, s0b = {}, s0c = {}, s0d = {};    // support 0
    v8f s1a = {}, s1b = {}, s1c = {}, s1d = {};    // support 1

    for (int it = 0; it < K / 16; ++it) {
        // A fragments: contiguous 8B loads, compile-time offsets
        v2f a0 = *reinterpret_cast<const v2f*>(ap + 0);
        v2f a1 = *reinterpret_cast<const v2f*>(ap + 4);
        v2f a2 = *reinterpret_cast<const v2f*>(ap + 8);
        v2f a3 = *reinterpret_cast<const v2f*>(ap + 12);
        // B fragments: compile-time strides -> immediate load offsets
        v2f b00, b01, b02, b03, b10, b11, b12, b13;
        b00.x = b0c[0 * Nout];   b00.y = b0c[1 * Nout];
        b01.x = b0c[4 * Nout];   b01.y = b0c[5 * Nout];
        b02.x = b0c[8 * Nout];   b02.y = b0c[9 * Nout];
        b03.x = b0c[12 * Nout];  b03.y = b0c[13 * Nout];
        b10.x = b1c[0 * Nout];   b10.y = b1c[1 * Nout];
        b11.x = b1c[4 * Nout];   b11.y = b1c[5 * Nout];
        b12.x = b1c[8 * Nout];   b12.y = b1c[9 * Nout];
        b13.x = b1c[12 * Nout];  b13.y = b1c[13 * Nout];

        s0a = __builtin_amdgcn_wmma_f32_16x16x4_f32(false, a0, false, b00, (short)0, s0a, false, false);
        s1a = __builtin_amdgcn_wmma_f32_16x16x4_f32(false, a0, false, b10, (short)0, s1a, false, false);
        s0b = __builtin_amdgcn_wmma_f32_16x16x4_f32(false, a1, false, b01, (short)0, s0b, false, false);
        s1b = __builtin_amdgcn_wmma_f32_16x16x4_f32(false, a1, false, b11, (short)0, s1b, false, false);
        s0c = __builtin_amdgcn_wmma_f32_16x16x4_f32(false, a2, false, b02, (short)0, s0c, false, false);
        s1c = __builtin_amdgcn_wmma_f32_16x16x4_f32(false, a2, false, b12, (short)0, s1c, false, false);
        s0d = __builtin_amdgcn_wmma_f32_16x16x4_f32(false, a3, false, b03, (short)0, s0d, false, false);
        s1d = __builtin_amdgcn_wmma_f32_16x16x4_f32(false, a3, false, b13, (short)0, s1d, false, false);

        ap  += 16;
        b0c += (size_t)16 * Nout;
        b1c += (size_t)16 * Nout;
    }
    v8f r0 = (s0a + s0b) + (s0c + s0d);
    v8f r1 = (s1a + s1b) + (s1c + s1d);

    const size_t coff = (size_t)(tileM * 16 + half * 8) * Nout + tileN * 16 + mn;
    float* __restrict__ c0row = C0 + coff;
    float* __restrict__ c1row = C1 + coff;
#pragma unroll
    for (int v = 0; v < 8; ++v) c0row[(size_t)v * Nout] = r0[v];
#pragma unroll
    for (int v = 0; v < 8; ++v) c1row[(size_t)v * Nout] = r1[v];
}

// ---------------------------------------------------------------------------
// Fused dual-support edge aggregation:
//   acc[row,:] += v0 * pre0[col,:] + v1 * pre1[col,:]
// Hout = 4 << LOGC (compile-time): chunk decomposition is shifts/masks only.
// One atomic RMW per destination element (halved vs per-support passes).
// Edge arrays streamed with non-temporal loads to preserve L2 residency of
// pre/acc (CDNA5 TH=NT policy).
// ---------------------------------------------------------------------------
template <int LOGC>
__global__ void gcn_scatter_fused(const float* __restrict__ pre0,
                                  const float* __restrict__ pre1,
                                  const int* __restrict__ erow,
                                  const int* __restrict__ ecol,
                                  const float* __restrict__ ev0,
                                  const float* __restrict__ ev1,
                                  float* __restrict__ acc) {
    constexpr int C = 1 << LOGC;                     // float4 chunks per row
    int t = blockIdx.x * blockDim.x + threadIdx.x;
    int e = t >> LOGC;
    int c = t & (C - 1);
    if (e >= EE) return;

    float v0 = __builtin_nontemporal_load(ev0 + e);
    float v1 = __builtin_nontemporal_load(ev1 + e);
    int   r  = __builtin_nontemporal_load(erow + e);
    int  col = __builtin_nontemporal_load(ecol + e);

    const float4 p0 =
        reinterpret_cast<const float4*>(pre0 + ((size_t)col << (LOGC + 2)))[c];
    const float4 p1 =
        reinterpret_cast<const float4*>(pre1 + ((size_t)col << (LOGC + 2)))[c];

    float* dst = acc + ((size_t)r << (LOGC + 2)) + 4 * c;
    atomicAdd(dst + 0, v0 * p0.x + v1 * p1.x);
    atomicAdd(dst + 1, v0 * p0.y + v1 * p1.y);
    atomicAdd(dst + 2, v0 * p0.z + v1 * p1.z);
    atomicAdd(dst + 3, v0 * p0.w + v1 * p1.w);
}

// fused scalar scatter for Hout == 1 (layer 3): one atomic per edge
__global__ void gcn_scatter1_fused(const float* __restrict__ p0,
                                   const float* __restrict__ p1,
                                   const int* __restrict__ erow,
                                   const int* __restrict__ ecol,
                                   const float* __restrict__ ev0,
                                   const float* __restrict__ ev1,
                                   float* __restrict__ acc) {
    int e = blockIdx.x * blockDim.x + threadIdx.x;
    if (e >= EE) return;
    float v0 = __builtin_nontemporal_load(ev0 + e);
    float v1 = __builtin_nontemporal_load(ev1 + e);
    int  col = __builtin_nontemporal_load(ecol + e);
    int  r   = __builtin_nontemporal_load(erow + e);
    atomicAdd(acc + r, v0 * p0[col] + v1 * p1[col]);
}

// layer-3 projection, both supports in one pass (h2 row read once):
//   p0[i] = dot(h2[i,:], w0), p1[i] = dot(h2[i,:], w1)
__global__ void gcn_dot64x2(const float* __restrict__ h,
                            const float* __restrict__ w0,
                            const float* __restrict__ w1,
                            float* __restrict__ p0, float* __restrict__ p1) {
    int i = blockIdx.x * blockDim.x + threadIdx.x;
    if (i >= NN) return;
    const float* row = h + (size_t)i * HH2;
    float s0 = 0.0f, s1 = 0.0f;
#pragma unroll
    for (int k = 0; k < HH2; ++k) {
        float hv = row[k];
        s0 += hv * w0[k];
        s1 += hv * w1[k];
    }
    p0[i] = s0;
    p1[i] = s1;
}

// ---------------------------------------------------------------------------
extern "C" void kernel_launch(void* const* d_in, const int* in_sizes, int n_in,
                              void* d_out, int out_size, void* d_ws, size_t ws_size,
                              hipStream_t stream) {
    const float* x     = (const float*)d_in[0];   // [N, 256]
    const int*   erow  = (const int*)d_in[1];     // [E]
    const int*   ecol  = (const int*)d_in[2];     // [E]
    const float* evals = (const float*)d_in[3];   // [S, E]
    const float* W1    = (const float*)d_in[4];   // [S, 256, 128]
    const float* W2    = (const float*)d_in[5];   // [S, 128, 64]
    const float* W3    = (const float*)d_in[6];   // [S, 64, 1]
    float* out = (float*)d_out;                   // [N, 1]

    const float* ev0 = evals;
    const float* ev1 = evals + (size_t)EE;

    // workspace layout (floats): pre0 | pre1 | bufA (h1) | bufB (h2)
    float* pre0 = (float*)d_ws;                     // N * 128
    float* pre1 = pre0 + (size_t)NN * HH1;          // N * 128
    float* bufA = pre1 + (size_t)NN * HH1;          // N * 128
    float* bufB = bufA + (size_t)NN * HH1;          // N * 64

    const int ZB = 256;
    dim3 gemm_blk(32, GEMM_WAVES);
    const int mtiles = NN / 16;                     // 6250 exactly
    const int gemm_gy = (mtiles + GEMM_WAVES - 1) / GEMM_WAVES;

    // ---------------- layer 1: x[N,256] @ W1 -> bufA[N,128], relu ----------
    gcn_zero<<<((size_t)NN * HH1 + ZB - 1) / ZB, ZB, 0, stream>>>(bufA, NN * HH1);
    gcn_gemm_wmma2<FF, HH1><<<dim3(HH1 / 16, gemm_gy), gemm_blk, 0, stream>>>(
        x, W1, W1 + (size_t)FF * HH1, pre0, pre1, NN);
    {
        long long tot = (long long)EE << 5;         // E * 32 chunks
        gcn_scatter_fused<5><<<(tot + ZB - 1) / ZB, ZB, 0, stream>>>(
            pre0, pre1, erow, ecol, ev0, ev1, bufA);
    }
    gcn_relu<<<((size_t)NN * HH1 + ZB - 1) / ZB, ZB, 0, stream>>>(bufA, NN * HH1);

    // ---------------- layer 2: bufA[N,128] @ W2 -> bufB[N,64], relu --------
    gcn_zero<<<((size_t)NN * HH2 + ZB - 1) / ZB, ZB, 0, stream>>>(bufB, NN * HH2);
    gcn_gemm_wmma2<HH1, HH2><<<dim3(HH2 / 16, gemm_gy), gemm_blk, 0, stream>>>(
        bufA, W2, W2 + (size_t)HH1 * HH2, pre0, pre1, NN);
    {
        long long tot = (long long)EE << 4;         // E * 16 chunks
        gcn_scatter_fused<4><<<(tot + ZB - 1) / ZB, ZB, 0, stream>>>(
            pre0, pre1, erow, ecol, ev0, ev1, bufB);
    }
    gcn_relu<<<((size_t)NN * HH2 + ZB - 1) / ZB, ZB, 0, stream>>>(bufB, NN * HH2);

    // ---------------- layer 3: bufB[N,64] @ W3 -> out[N,1] (no act) --------
    gcn_zero<<<(NN + ZB - 1) / ZB, ZB, 0, stream>>>(out, NN);
    gcn_dot64x2<<<(NN + ZB - 1) / ZB, ZB, 0, stream>>>(
        bufB, W3, W3 + (size_t)HH2, pre0, pre1);
    gcn_scatter1_fused<<<(EE + ZB - 1) / ZB, ZB, 0, stream>>>(
        pre0, pre1, erow, ecol, ev0, ev1, out);
}